// RQAE_47115791237179
// MI455X (gfx1250) — compile-verified
//
#include <hip/hip_runtime.h>

typedef float v2f __attribute__((ext_vector_type(2)));
typedef float v8f __attribute__((ext_vector_type(8)));

#define RQ_DIM     2304
#define RQ_RSTRIDE 2308   // padded row stride (floats): stride%64==4 -> conflict-free column access
#define RQ_NQ      1024
#define RQ_NCB     625
#define RQ_NCBP    640    // codebook padded to 40 tiles of 16 (pad rows are zero, can only tie, never win)
#define RQ_TOK     16     // tokens per workgroup
#define RQ_NTOK    8192   // B*S
#define RQ_THREADS 128    // 4 waves of 32

// LDS layout (float offsets) -- total kept <= 160KB so 2 workgroups/WGP can co-reside
#define SM_R     0
#define SM_CB    (RQ_TOK * RQ_RSTRIDE)        // 36928
#define SM_QP    (SM_CB + RQ_NCBP * 4)        // 39488 : per-wave partial q [4][16][4]
#define SM_Q     (SM_QP + 4 * 16 * 4)         // 39744 : q [16][4]
#define SM_CV    (SM_Q + 64)                  // 39808 : chosen codevectors [16][4]
#define SM_RED   (SM_CV + 64)                 // 39872 : argmax candidates [4][16][2] (val, idx)
#define SM_TOTAL (SM_RED + 128)               // 40000 floats = 160000 bytes

__global__ __launch_bounds__(RQ_THREADS, 1)
void rqae_kernel(const float* __restrict__ x,
                 const float* __restrict__ W_in,   // [NQ][4][2304]
                 const float* __restrict__ b_in,   // [NQ][4]
                 const float* __restrict__ W_out,  // [NQ][2304][4]
                 const float* __restrict__ b_out,  // [NQ][2304]
                 const float* __restrict__ cbg,    // [NQ][625][4] (all stages identical; use stage 0)
                 float* __restrict__ qout,         // [8192][2304]
                 float* __restrict__ idxout)       // [8192][1024] (indices as float)
{
    extern __shared__ float sm[];
    float* R   = sm + SM_R;
    float* CB  = sm + SM_CB;
    float* QP  = sm + SM_QP;
    float* Q   = sm + SM_Q;
    float* CV  = sm + SM_CV;
    float* RED = sm + SM_RED;
    int*   REDI = (int*)RED;

    const int tid  = threadIdx.x;
    const int lane = tid & 31;
    const int wave = tid >> 5;          // 0..3
    const int h    = lane >> 4;         // lane-half: 0/1
    const int n16  = lane & 15;
    const long tok0 = (long)blockIdx.x * RQ_TOK;

    // ---- init: x tile -> LDS residual (padded rows); codebook -> LDS (zero padded) ----
    for (int i = tid; i < RQ_TOK * RQ_DIM; i += RQ_THREADS) {
        int t = i / RQ_DIM;
        int d = i - t * RQ_DIM;
        R[t * RQ_RSTRIDE + d] = x[(tok0 + t) * RQ_DIM + d];
    }
    for (int i = tid; i < RQ_NCBP * 4; i += RQ_THREADS) {
        CB[i] = (i < RQ_NCB * 4) ? cbg[i] : 0.0f;
    }
    __syncthreads();

    // hoisted per-lane bases
    const int   wrow  = (n16 < 4) ? n16 : 3;          // clamped W_in row: cols 4..15 of C are never read,
                                                      // so garbage B columns are harmless (no masking!)
    const float* Rrow = R  + n16 * RQ_RSTRIDE + 2 * h;   // A: M=n16, K = 2h, 2h+1
    const float* CBl  = CB + n16 * 4 + 2 * h;            // B base for codebook tiles
    const int   kbase = wave * (RQ_DIM / 4);             // 576 floats of K per wave

    for (int n = 0; n < RQ_NQ; ++n) {
        const float* Wi = W_in  + (long)n * (4 * RQ_DIM);
        const float* Bi = b_in  + (long)n * 4;
        const float* Wo = W_out + (long)n * (RQ_DIM * 4);
        const float* Bo = b_out + (long)n * RQ_DIM;

        // hint next stage's weights toward the caches (global_prefetch_b8)
        if (n + 1 < RQ_NQ) {
            __builtin_prefetch(Wi + 4 * RQ_DIM + tid * 72, 0, 0);
            __builtin_prefetch(Wo + RQ_DIM * 4 + tid * 72, 0, 0);
            __builtin_prefetch(Bo + RQ_DIM + tid * 18, 0, 0);
        }

        // ================= lin_in: q[16x4] = R[16x2304] @ Wi^T =================
        // K split 4 ways across waves; V_WMMA_F32_16X16X4_F32, N padded 4->16.
        // Software pipeline: operands for iteration i+1 are issued before the
        // 4 WMMAs of iteration i, hiding LDS/L2 latency behind matrix work.
        v8f acc0 = {0.f,0.f,0.f,0.f,0.f,0.f,0.f,0.f};
        v8f acc1 = {0.f,0.f,0.f,0.f,0.f,0.f,0.f,0.f};
        const float* Wrow = Wi + wrow * RQ_DIM + 2 * h;
        v2f pa[4], pb[4];
        #pragma unroll
        for (int u = 0; u < 4; ++u) {
            pa[u] = *(const v2f*)(Rrow + kbase + 4 * u);
            pb[u] = *(const v2f*)(Wrow + kbase + 4 * u);
        }
        for (int c = 0; c < RQ_DIM / 16; c += 4) {       // 144 chunks of K=4, 4 per iteration
            v2f ca[4], cb4[4];
            #pragma unroll
            for (int u = 0; u < 4; ++u) { ca[u] = pa[u]; cb4[u] = pb[u]; }
            const int cn = (c + 4 < RQ_DIM / 16) ? (c + 4) : 0;  // branchless wrap: no OOB on last iter
            #pragma unroll
            for (int u = 0; u < 4; ++u) {
                pa[u] = *(const v2f*)(Rrow + kbase + 4 * (cn + u));
                pb[u] = *(const v2f*)(Wrow + kbase + 4 * (cn + u));
            }
            acc0 = __builtin_amdgcn_wmma_f32_16x16x4_f32(false, ca[0], false, cb4[0], (short)0, acc0, false, false);
            acc1 = __builtin_amdgcn_wmma_f32_16x16x4_f32(false, ca[1], false, cb4[1], (short)0, acc1, false, false);
            acc0 = __builtin_amdgcn_wmma_f32_16x16x4_f32(false, ca[2], false, cb4[2], (short)0, acc0, false, false);
            acc1 = __builtin_amdgcn_wmma_f32_16x16x4_f32(false, ca[3], false, cb4[3], (short)0, acc1, false, false);
        }
        v8f acc = acc0 + acc1;
        // C layout: vgpr r, lane: M = r + 8h, N = n16. Extract columns 0..3 only.
        if (n16 < 4) {
            #pragma unroll
            for (int r = 0; r < 8; ++r)
                QP[(wave * 16 + r + 8 * h) * 4 + n16] = acc[r];
        }
        __syncthreads();
        // combine 4 wave-partials + bias -> Q[16][4]
        if (tid < 64) {
            int t = tid >> 2, c4 = tid & 3;
            float v = Bi[c4];
            #pragma unroll
            for (int w = 0; w < 4; ++w) v += QP[(w * 16 + t) * 4 + c4];
            Q[t * 4 + c4] = v;
        }
        __syncthreads();

        // ========== codebook argmax (cosine argmax == raw dot argmax) ==========
        v2f aq = *(const v2f*)(Q + n16 * 4 + 2 * h);
        float best[8]; int bidx[8];
        #pragma unroll
        for (int r = 0; r < 8; ++r) { best[r] = -__builtin_inff(); bidx[r] = 0; }
        for (int ct = wave * 10; ct < wave * 10 + 10; ++ct) {   // 40 tiles / 4 waves
            v2f b = *(const v2f*)(CBl + ct * 64);
            v8f cc = {0.f,0.f,0.f,0.f,0.f,0.f,0.f,0.f};
            cc = __builtin_amdgcn_wmma_f32_16x16x4_f32(false, aq, false, b, (short)0, cc, false, false);
            const int idx = ct * 16 + n16;
            #pragma unroll
            for (int r = 0; r < 8; ++r)
                if (cc[r] > best[r]) { best[r] = cc[r]; bidx[r] = idx; } // strict > keeps first max
        }
        // reduce across the 16 lanes of each half (xor masks < 16 stay in-half)
        #pragma unroll
        for (int m = 1; m < 16; m <<= 1) {
            #pragma unroll
            for (int r = 0; r < 8; ++r) {
                float ov = __shfl_xor(best[r], m, 32);
                int   oi = __shfl_xor(bidx[r], m, 32);
                if (ov > best[r] || (ov == best[r] && oi < bidx[r])) { best[r] = ov; bidx[r] = oi; }
            }
        }
        if (n16 == 0) {
            #pragma unroll
            for (int r = 0; r < 8; ++r) {
                int t = r + 8 * h;
                RED [(wave * 16 + t) * 2 + 0] = best[r];
                REDI[(wave * 16 + t) * 2 + 1] = bidx[r];
            }
        }
        __syncthreads();
        // final per-token selection across the 4 waves; emit index + codevector
        if (tid < 16) {
            float bv = -__builtin_inff(); int bi2 = 0x7fffffff;
            #pragma unroll
            for (int w = 0; w < 4; ++w) {
                float v = RED [(w * 16 + tid) * 2 + 0];
                int   i = REDI[(w * 16 + tid) * 2 + 1];
                if (v > bv || (v == bv && i < bi2)) { bv = v; bi2 = i; }
            }
            idxout[(tok0 + tid) * RQ_NQ + n] = (float)bi2;
            CV[tid * 4 + 0] = CB[bi2 * 4 + 0];
            CV[tid * 4 + 1] = CB[bi2 * 4 + 1];
            CV[tid * 4 + 2] = CB[bi2 * 4 + 2];
            CV[tid * 4 + 3] = CB[bi2 * 4 + 3];
        }
        __syncthreads();

        // ========= lin_out: out[16x2304] = cv[16x4] @ Wo^T + Bo; R -= out =========
        // 1-ahead pipeline: tile j's ds_add work covers tile j+1's global loads.
        v2f av = *(const v2f*)(CV + n16 * 4 + 2 * h);
        const float* Wop = Wo + n16 * 4 + 2 * h;
        {
            const int d0 = (wave * 36) * 16;
            float bo_n = Bo[d0 + n16];
            v2f  b_n  = *(const v2f*)(Wop + d0 * 4);
            for (int j = 0; j < 36; ++j) {               // 144 N-tiles / 4 waves
                const float bo = bo_n;
                const v2f   b  = b_n;
                const int jn = (j + 1 < 36) ? (j + 1) : 0;   // branchless wrap
                const int d1 = (wave * 36 + jn) * 16;
                bo_n = Bo[d1 + n16];
                b_n  = *(const v2f*)(Wop + d1 * 4);
                v8f cc = {bo,bo,bo,bo,bo,bo,bo,bo};
                cc = __builtin_amdgcn_wmma_f32_16x16x4_f32(false, av, false, b, (short)0, cc, false, false);
                const int dcur = (wave * 36 + j) * 16;
                float* rr = R + 8 * h * RQ_RSTRIDE + dcur + n16;
                #pragma unroll
                for (int r = 0; r < 8; ++r)              // each (token,d) owned by exactly one lane:
                    __hip_atomic_fetch_add(&rr[r * RQ_RSTRIDE], -cc[r],   // ds_add_f32, no return
                                           __ATOMIC_RELAXED, __HIP_MEMORY_SCOPE_WORKGROUP);
            }
        }
        __syncthreads();
    }

    // quantized_out = x - final residual
    for (int i = tid; i < RQ_TOK * RQ_DIM; i += RQ_THREADS) {
        int t = i / RQ_DIM;
        int d = i - t * RQ_DIM;
        qout[(tok0 + t) * RQ_DIM + d] = x[(tok0 + t) * RQ_DIM + d] - R[t * RQ_RSTRIDE + d];
    }
}

extern "C" void kernel_launch(void* const* d_in, const int* in_sizes, int n_in,
                              void* d_out, int out_size, void* d_ws, size_t ws_size,
                              hipStream_t stream) {
    (void)in_sizes; (void)n_in; (void)out_size; (void)d_ws; (void)ws_size;
    const float* x     = (const float*)d_in[0];
    const float* W_in  = (const float*)d_in[1];
    const float* b_in  = (const float*)d_in[2];
    const float* W_out = (const float*)d_in[3];
    const float* b_out = (const float*)d_in[4];
    const float* cb    = (const float*)d_in[5];
    float* qout = (float*)d_out;
    float* idxf = qout + (size_t)RQ_NTOK * RQ_DIM;   // outputs concatenated: [8192*2304] then [8192*1024]

    dim3 grid(RQ_NTOK / RQ_TOK);   // 512 workgroups, residual tile resident in LDS
    dim3 block(RQ_THREADS);        // 4 waves (wave32)
    size_t smem = (size_t)SM_TOTAL * sizeof(float);  // 160000 B of the 320KB/WGP LDS
    rqae_kernel<<<grid, block, smem, stream>>>(x, W_in, b_in, W_out, b_out, cb, qout, idxf);
}